// JointBacksolveAdjoint_22548578304083
// MI455X (gfx1250) — compile-verified
//
#include <hip/hip_runtime.h>
#include <hip/hip_bf16.h>
#include <math.h>

typedef float v2f  __attribute__((ext_vector_type(2)));
typedef float v4fv __attribute__((ext_vector_type(4)));
typedef float v8f  __attribute__((ext_vector_type(8)));
typedef int   v4i  __attribute__((ext_vector_type(4)));

#define Bn 128
#define Fn 1024
#define Hn 2048
#define NSTEPS 48
#define DT0 0.0625f

// ---- async global->LDS copy (gfx1250). Fallback: load+ds_store. -----------
#if __has_builtin(__builtin_amdgcn_global_load_async_to_lds_b128)
#define HAVE_ASYNC_LDS 1
#define COPY16(d, s) __builtin_amdgcn_global_load_async_to_lds_b128( \
    (__attribute__((address_space(1))) v4i*)(v4fv*)(s),              \
    (__attribute__((address_space(3))) v4i*)(d), 0, 0)
static __device__ __forceinline__ void wait_async() {
#if __has_builtin(__builtin_amdgcn_s_wait_asynccnt)
  __builtin_amdgcn_s_wait_asynccnt(0);
#else
  asm volatile("s_wait_asynccnt 0x0" ::: "memory");
#endif
}
#else
#define COPY16(d, s) (*(d) = *(s))
static __device__ __forceinline__ void wait_async() {}
#endif

// Dormand-Prince coefficients (f32-rounded exactly like the reference)
__constant__ float cA[7][6] = {
  {0.f,0.f,0.f,0.f,0.f,0.f},
  {(float)(1.0/5.0),0.f,0.f,0.f,0.f,0.f},
  {(float)(3.0/40.0),(float)(9.0/40.0),0.f,0.f,0.f,0.f},
  {(float)(44.0/45.0),(float)(-56.0/15.0),(float)(32.0/9.0),0.f,0.f,0.f},
  {(float)(19372.0/6561.0),(float)(-25360.0/2187.0),(float)(64448.0/6561.0),(float)(-212.0/729.0),0.f,0.f},
  {(float)(9017.0/3168.0),(float)(-355.0/33.0),(float)(46732.0/5247.0),(float)(49.0/176.0),(float)(-5103.0/18656.0),0.f},
  {(float)(35.0/384.0),0.f,(float)(500.0/1113.0),(float)(125.0/192.0),(float)(-2187.0/6784.0),(float)(11.0/84.0)}
};
__constant__ float cB5[7] = {
  (float)(35.0/384.0),0.f,(float)(500.0/1113.0),(float)(125.0/192.0),
  (float)(-2187.0/6784.0),(float)(11.0/84.0),0.f
};
__constant__ float cE[7] = {
  (float)(35.0/384.0)  - (float)(5179.0/57600.0),
  0.f,
  (float)(500.0/1113.0) - (float)(7571.0/16695.0),
  (float)(125.0/192.0)  - (float)(393.0/640.0),
  (float)(-2187.0/6784.0) + (float)(92097.0/339200.0),
  (float)(11.0/84.0)    - (float)(187.0/2100.0),
  -(float)(1.0/40.0)
};

// ---------------------------------------------------------------------------
// Repack row-major W[K][N] into WMMA B-fragment layout:
//   Wp[(s*Ntiles + nt)*32 + lane] = {W[k][n], W[k+1][n]},
//   k = 4s + 2*(lane>>4), n = 16*nt + (lane&15)
// ---------------------------------------------------------------------------
__global__ __launch_bounds__(256) void pack_w(const float* __restrict__ W,
                                              v2f* __restrict__ Wp,
                                              int K, int N, int logNtiles)
{
  int p = blockIdx.x * blockDim.x + threadIdx.x;
  int total = (K * N) >> 1;
  if (p >= total) return;
  int l  = p & 31;
  int q  = p >> 5;
  int nt = q & ((1 << logNtiles) - 1);
  int s  = q >> logNtiles;
  int k  = 4 * s + ((l >> 4) << 1);
  int n  = nt * 16 + (l & 15);
  v2f v;
  v.x = W[(size_t)k * N + n];
  v.y = W[(size_t)(k + 1) * N + n];
  Wp[p] = v;
}

// ---------------------------------------------------------------------------
// Stage input yi = y + dte * sum_j A[stage][j]*k_j, written in WMMA
// A-fragment layout: Apack[(mtile*Ksteps + s)*32 + lane].
// ---------------------------------------------------------------------------
__global__ __launch_bounds__(256) void stage_prep(const float* __restrict__ y,
                                                  const float* __restrict__ ks,
                                                  const float* __restrict__ dte,
                                                  v2f* __restrict__ Apack,
                                                  int stage)
{
  int p = blockIdx.x * blockDim.x + threadIdx.x;   // over B*F/2 float2s
  if (p >= Bn * Fn / 2) return;
  int l     = p & 31;
  int q     = p >> 5;          // mtile*(F/4) + s ; F/4 = 256
  int s     = q & 255;
  int mtile = q >> 8;
  int k     = 4 * s + ((l >> 4) << 1);
  int m     = mtile * 16 + (l & 15);
  size_t idx = (size_t)m * Fn + k;

  v2f v = *(const v2f*)(y + idx);
  float d = dte[m];
  for (int j = 0; j < stage; ++j) {
    float a = cA[stage][j];
    v2f kj = *(const v2f*)(ks + (size_t)j * Bn * Fn + idx);
    v.x += d * (a * kj.x);
    v.y += d * (a * kj.y);
  }
  Apack[p] = v;
}

// ---------------------------------------------------------------------------
// WMMA fp32 GEMM, LDS double-buffered:
//   block = 256 threads = 8 waves, computes 32 rows x 128 cols.
//   wave w: mtile-in-block = w>>2, 32-col strip = w&3 (two 16x16 accumulators)
//   A and B fragments staged through LDS (async copies) in chunks of 8 k-steps.
// TANH:     fused tanh (layer 1)
// PACK_OUT: scatter output directly into WMMA A-fragment layout (feeds GEMM2)
// ---------------------------------------------------------------------------
template <bool TANH, bool PACK_OUT>
__global__ __launch_bounds__(256) void gemm_wmma(const v2f* __restrict__ Apack,
                                                 const v2f* __restrict__ Wpack,
                                                 const float* __restrict__ bias,
                                                 float* __restrict__ Out,
                                                 int K, int N)
{
  constexpr int CH = 8;                       // k-steps per chunk (K granule = 32)
  __shared__ __align__(16) v2f sA[2][2][CH][32];   //  8 KB
  __shared__ __align__(16) v2f sB[2][CH][8][32];   // 32 KB

  const int t      = threadIdx.x;
  const int lane   = t & 31;
  const int w      = t >> 5;
  const int mt2    = w >> 2;          // 0..1
  const int strip  = w & 3;           // 0..3
  const int mtile0 = blockIdx.y * 2;
  const int nt0    = blockIdx.x * 8;  // first 16-col tile of this block
  const int Ksteps = K >> 2;
  const int Ntiles = N >> 4;

  auto stage_chunk = [&](int buf, int s0) {
    // A: 4 KB (2 mtiles x CH steps x 32 lanes x 8B) in one 256-thread pass
    {
      int mtc = t >> 7, ia = t & 127;
      const v4fv* src =
          (const v4fv*)(Apack + ((size_t)(mtile0 + mtc) * Ksteps + s0) * 32);
      COPY16(((v4fv*)&sA[buf][0][0][0]) + t, src + ia);
    }
    // B: 16 KB (CH steps x 8 ntiles x 32 lanes x 8B) in 4 passes
#pragma unroll
    for (int i = 0; i < 4; ++i) {
      int s  = i * 2 + (t >> 7);
      int ib = t & 127;
      const v4fv* src =
          (const v4fv*)(Wpack + ((size_t)(s0 + s) * Ntiles + nt0) * 32);
      COPY16(((v4fv*)&sB[buf][s][0][0]) + ib, src + ib);
    }
  };

  v8f c0 = {};
  v8f c1 = {};
  const int nchunks = Ksteps / CH;

  stage_chunk(0, 0);
  wait_async();
  __syncthreads();

  for (int c = 0; c < nchunks; ++c) {
    int buf = c & 1;
    if (c + 1 < nchunks) stage_chunk(buf ^ 1, (c + 1) * CH);
#pragma unroll
    for (int s = 0; s < CH; ++s) {
      v2f a  = sA[buf][mt2][s][lane];
      v2f b0 = sB[buf][s][strip * 2 + 0][lane];
      v2f b1 = sB[buf][s][strip * 2 + 1][lane];
      c0 = __builtin_amdgcn_wmma_f32_16x16x4_f32(false, a, false, b0, (short)0, c0, false, false);
      c1 = __builtin_amdgcn_wmma_f32_16x16x4_f32(false, a, false, b1, (short)0, c1, false, false);
    }
    wait_async();
    __syncthreads();
  }

  // Epilogue. C layout: VGPR r, lanes 0-15 -> M=r ; lanes 16-31 -> M=r+8 ; N=lane&15.
  const int mtile = mtile0 + mt2;
  const int n0    = blockIdx.x * 128 + strip * 32;
  const int mrow0 = mtile * 16 + ((lane >> 4) << 3);
  const int ncol0 = n0 + (lane & 15);
#pragma unroll
  for (int r = 0; r < 8; ++r) {
    int m = mrow0 + r;
#pragma unroll
    for (int sub = 0; sub < 2; ++sub) {
      int n = ncol0 + sub * 16;
      float v = (sub == 0 ? c0[r] : c1[r]) + bias[n];
      if (TANH) v = tanhf(v);
      if (PACK_OUT) {
        int s2   = n >> 2;
        int l2   = (m & 15) | ((n & 2) << 3);
        int comp = n & 1;
        Out[(((size_t)(m >> 4) * (N >> 2) + s2) * 32 + l2) * 2 + comp] = v;
      } else {
        Out[(size_t)m * N + n] = v;
      }
    }
  }
}

// ---------------------------------------------------------------------------
__global__ __launch_bounds__(256) void init_state(const float* __restrict__ y0,
                                                  const float* __restrict__ t_start,
                                                  float* __restrict__ t,
                                                  float* __restrict__ y,
                                                  float* __restrict__ dt,
                                                  int* __restrict__ running,
                                                  int* __restrict__ nsteps,
                                                  int* __restrict__ nacc)
{
  int i = blockIdx.x * blockDim.x + threadIdx.x;
  if (i < Bn * Fn) y[i] = y0[i];
  if (i < Bn) {
    t[i] = t_start[i];
    dt[i] = DT0;
    running[i] = 1;
    nsteps[i] = 0;
    nacc[i] = 0;
  }
}

__global__ void begin_step(const float* __restrict__ t,
                           const float* __restrict__ dt,
                           const float* __restrict__ t_end,
                           float* __restrict__ dte)
{
  int b = threadIdx.x;
  if (b < Bn) dte[b] = fminf(dt[b], t_end[b] - t[b]);
}

// ---------------------------------------------------------------------------
__global__ __launch_bounds__(256) void step_update(float* __restrict__ t,
                                                   float* __restrict__ y,
                                                   float* __restrict__ dt,
                                                   int* __restrict__ running,
                                                   int* __restrict__ nsteps,
                                                   int* __restrict__ nacc,
                                                   const float* __restrict__ dte,
                                                   const float* __restrict__ ks,
                                                   const float* __restrict__ t_end)
{
  const int b   = blockIdx.x;
  const int tid = threadIdx.x;
  const float d = dte[b];

  float ynew[4];
  float acc = 0.f;
#pragma unroll
  for (int j = 0; j < 4; ++j) {
    int e = tid + j * 256;
    size_t idx = (size_t)b * Fn + e;
    float yv = y[idx];
    float sb = 0.f, se = 0.f;
#pragma unroll
    for (int i = 0; i < 7; ++i) {
      float kv = ks[(size_t)i * Bn * Fn + idx];
      sb += cB5[i] * kv;
      se += cE[i] * kv;
    }
    float yn  = yv + d * sb;
    float err = d * se;
    float scale = 1e-6f + 1e-5f * fmaxf(fabsf(yv), fabsf(yn));
    float r = err / scale;
    acc += r * r;
    ynew[j] = yn;
  }

  __shared__ float red[256];
  __shared__ int s_toupd;
  red[tid] = acc;
  __syncthreads();
  for (int off = 128; off > 0; off >>= 1) {
    if (tid < off) red[tid] += red[tid + off];
    __syncthreads();
  }
  if (tid == 0) {
    float norm = sqrtf(red[0] / (float)Fn);
    int accept  = norm <= 1.0f;
    int run_old = running[b];
    int to_upd  = accept && run_old;
    float tn = to_upd ? (t[b] + d) : t[b];
    nsteps[b] += run_old ? 1 : 0;
    nacc[b]   += to_upd ? 1 : 0;
    float factor = fminf(fmaxf(0.9f * powf(fmaxf(norm, 1e-10f), -0.2f), 0.2f), 10.0f);
    int run_new = run_old && (tn < t_end[b]);
    float dtn = dt[b] * factor;
    if (run_new) dt[b] = dtn;
    t[b] = tn;
    running[b] = run_new;
    s_toupd = to_upd;
  }
  __syncthreads();
  if (s_toupd) {
#pragma unroll
    for (int j = 0; j < 4; ++j)
      y[(size_t)b * Fn + tid + j * 256] = ynew[j];
  }
}

__global__ __launch_bounds__(256) void finalize(const float* __restrict__ y,
                                                const int* __restrict__ nsteps,
                                                const int* __restrict__ nacc,
                                                float* __restrict__ out)
{
  int i = blockIdx.x * blockDim.x + threadIdx.x;
  if (i < Bn * Fn) out[i] = y[i];
  if (i < Bn) {
    int* oi = (int*)out;
    oi[Bn * Fn + i]      = nsteps[i];
    oi[Bn * Fn + Bn + i] = nacc[i];
  }
}

// ---------------------------------------------------------------------------
extern "C" void kernel_launch(void* const* d_in, const int* in_sizes, int n_in,
                              void* d_out, int out_size, void* d_ws, size_t ws_size,
                              hipStream_t stream) {
  (void)in_sizes; (void)n_in; (void)out_size; (void)ws_size;
  const float* y0      = (const float*)d_in[0];
  const float* t_start = (const float*)d_in[1];
  const float* t_end   = (const float*)d_in[2];
  const float* W1      = (const float*)d_in[3];
  const float* b1      = (const float*)d_in[4];
  const float* W2      = (const float*)d_in[5];
  const float* b2      = (const float*)d_in[6];

  float* ws = (float*)d_ws;
  float* t       = ws;
  float* dt      = ws + Bn;
  float* dte     = ws + 2 * Bn;
  int*   running = (int*)(ws + 3 * Bn);
  int*   nsteps  = (int*)(ws + 4 * Bn);
  int*   nacc    = (int*)(ws + 5 * Bn);
  float* y    = ws + 1024;                       // [B,F] row-major
  float* yi   = y   + (size_t)Bn * Fn;           // [B,F] A-fragment packed
  float* h    = yi  + (size_t)Bn * Fn;           // [B,H] A-fragment packed
  float* ks   = h   + (size_t)Bn * Hn;           // 7 x [B,F] row-major
  float* Wp1  = ks  + (size_t)7 * Bn * Fn;       // [F,H] B-fragment packed
  float* Wp2  = Wp1 + (size_t)Fn * Hn;           // [H,F] B-fragment packed

  pack_w<<<(Fn * Hn / 2 + 255) / 256, 256, 0, stream>>>(W1, (v2f*)Wp1, Fn, Hn, 7); // Ntiles=128
  pack_w<<<(Hn * Fn / 2 + 255) / 256, 256, 0, stream>>>(W2, (v2f*)Wp2, Hn, Fn, 6); // Ntiles=64

  init_state<<<(Bn * Fn + 255) / 256, 256, 0, stream>>>(y0, t_start, t, y, dt, running, nsteps, nacc);

  for (int step = 0; step < NSTEPS; ++step) {
    begin_step<<<1, 128, 0, stream>>>(t, dt, t_end, dte);
    for (int i = 0; i < 7; ++i) {
      stage_prep<<<(Bn * Fn / 2 + 255) / 256, 256, 0, stream>>>(y, ks, dte, (v2f*)yi, i);
      gemm_wmma<true, true><<<dim3(Hn / 128, Bn / 32), 256, 0, stream>>>(
          (const v2f*)yi, (const v2f*)Wp1, b1, h, Fn, Hn);
      gemm_wmma<false, false><<<dim3(Fn / 128, Bn / 32), 256, 0, stream>>>(
          (const v2f*)h, (const v2f*)Wp2, b2, ks + (size_t)i * Bn * Fn, Hn, Fn);
    }
    step_update<<<Bn, 256, 0, stream>>>(t, y, dt, running, nsteps, nacc, dte, ks, t_end);
  }

  finalize<<<(Bn * Fn + 255) / 256, 256, 0, stream>>>(y, nsteps, nacc, (float*)d_out);
}